// Decoder_60722247630852
// MI455X (gfx1250) — compile-verified
//
#include <hip/hip_runtime.h>
#include <math.h>
#include <stdint.h>

// Problem dims (fixed by the reference)
#define Bsz   128
#define Kk    49
#define Dd    512
#define Ee    256
#define Vv    10000
#define Tt    20
#define FourD 2048   // 4*D
#define EDsum 768    // E+D

typedef __attribute__((ext_vector_type(2))) float v2f;
typedef __attribute__((ext_vector_type(4))) float v4f;
typedef __attribute__((ext_vector_type(8))) float v8f;
typedef __attribute__((ext_vector_type(4))) int   v4i;

// Address-space-qualified element types (AS1 = global/"__device__", AS3 = LDS)
typedef __attribute__((address_space(1))) v4i  gv4i;
typedef __attribute__((address_space(3))) v4i  lv4i;
typedef __attribute__((address_space(1))) void gvoid;
typedef __attribute__((address_space(3))) void lvoid;

#define SASTR 68     // LDS row stride (floats): multiple of 4 for aligned b128
                     // stores, and (4*row + col) % 64 stays conflict-free.

#if defined(__gfx1250__) && __has_builtin(__builtin_amdgcn_global_load_async_to_lds_b128)
#define HAS_ASYNC_LDS 1
#else
#define HAS_ASYNC_LDS 0
#endif

// Async global->LDS b128 copy (CDNA5 GLOBAL_LOAD_ASYNC_TO_LDS_B128, ASYNCcnt),
// falling back to a synchronous copy through VGPRs.
__device__ __forceinline__ void copy_b128_async(const float* g, float* l) {
#if HAS_ASYNC_LDS
    __builtin_amdgcn_global_load_async_to_lds_b128(
        (gv4i*)(gvoid*)(void*)(uintptr_t)g,
        (lv4i*)(lvoid*)(void*)l,
        0, 0);
#else
    *(v4f*)l = *(const v4f*)g;
#endif
}

template <int CNT>
__device__ __forceinline__ void wait_asynccnt_le() {
#if HAS_ASYNC_LDS
#if __has_builtin(__builtin_amdgcn_s_wait_asynccnt)
    __builtin_amdgcn_s_wait_asynccnt(CNT);
#else
    asm volatile("s_wait_asynccnt %0" :: "i"(CNT) : "memory");
#endif
#endif
    asm volatile("" ::: "memory");   // keep LDS reads below the wait
}

__device__ __forceinline__ float sigmoidf_(float x) {
    return 1.0f / (1.0f + __expf(-x));
}

// ---------------------------------------------------------------------------
// Generic fp32 WMMA GEMM:  out[m,n] = sum_k A[m,k]*W[n,k]  (+bias +bias2 +add)
// optionally masked to 0 where lengths[m] <= t.
// One wave computes a 16x64 output strip with V_WMMA_F32_16X16X4_F32 (4
// accumulators share each A fragment -> 2.5 LDS floats per WMMA).  A (16x64)
// and W (64x64) K-chunks are double-buffered in LDS and filled with async
// global->LDS b128 copies (40 per chunk), so chunk i+1's HBM/L2 latency
// overlaps chunk i's WMMA chain; s_wait_asynccnt gates consumption.
// Requires: M%16==0 (grid.y), N%16==0, Kd%64==0, A rows 16B-aligned.
// ---------------------------------------------------------------------------
__global__ __launch_bounds__(32)
void wmma_gemm_f32(const float* __restrict__ A, int lda,
                   const float* __restrict__ W, int ldw,
                   const float* __restrict__ bias,   // nullable
                   const float* __restrict__ bias2,  // nullable
                   const float* __restrict__ add, int ldadd,  // nullable
                   float* __restrict__ out, int ldc,
                   int Kd, int N,
                   const int* __restrict__ lengths, int t)    // nullable mask
{
    __shared__ float sA[2][16 * SASTR];
    __shared__ float sW[2][64 * SASTR];

    const int lane = threadIdx.x;          // 0..31
    const int m0   = blockIdx.y << 4;
    const int n0   = blockIdx.x << 6;      // 64-wide N strip
    const int half = lane >> 4;            // 0/1
    const int l16  = lane & 15;
    const int c4   = l16 * 4;              // 4 floats per lane per row

    // Issue one K-chunk's staging: 8 (A) + 32 (W) async b128 ops per wave.
    auto issue_chunk = [&](int k0, int buf) {
        #pragma unroll
        for (int r = 0; r < 16; r += 2) {
            const int rr = r + half;
            copy_b128_async(A + (m0 + rr) * lda + k0 + c4,
                            &sA[buf][rr * SASTR + c4]);
        }
        #pragma unroll 4
        for (int r = 0; r < 64; r += 2) {
            const int rr = r + half;
            int wr = n0 + rr;
            if (wr >= N) wr = N - 1;       // ragged N edge: in-bounds, discarded later
            copy_b128_async(W + wr * ldw + k0 + c4,
                            &sW[buf][rr * SASTR + c4]);
        }
    };

    v8f acc[4] = {v8f{}, v8f{}, v8f{}, v8f{}};

    const int nchunks = Kd >> 6;
    issue_chunk(0, 0);
    for (int ci = 0; ci < nchunks; ++ci) {
        const int buf = ci & 1;
        if (ci + 1 < nchunks) {
            issue_chunk((ci + 1) << 6, buf ^ 1);   // prefetch next chunk
            wait_asynccnt_le<40>();                // current chunk resident
        } else {
            wait_asynccnt_le<0>();
        }

        // Fragment layout (32-bit 16x4 A / 4x16 B):
        //   lanes 0-15: v0=K+0, v1=K+1 ; lanes 16-31: v0=K+2, v1=K+3
        const int kh = half << 1;
        const float* __restrict__ pa = &sA[buf][l16 * SASTR + kh];
        const float* __restrict__ pw = &sW[buf][l16 * SASTR + kh];
        #pragma unroll
        for (int kk = 0; kk < 64; kk += 4) {
            v2f a;
            a.x = pa[kk];
            a.y = pa[kk + 1];
            #pragma unroll
            for (int j = 0; j < 4; ++j) {
                v2f b;
                b.x = pw[j * 16 * SASTR + kk];
                b.y = pw[j * 16 * SASTR + kk + 1];
                acc[j] = __builtin_amdgcn_wmma_f32_16x16x4_f32(
                    false, a, false, b, (short)0, acc[j], false, false);
            }
        }
    }

    // Epilogue. C/D layout: VGPR i -> M = i + (lane>=16 ? 8 : 0), N = lane&15.
    const int rb = half << 3;
    #pragma unroll
    for (int j = 0; j < 4; ++j) {
        const int nj = n0 + j * 16;
        if (nj < N) {                       // wave-uniform guard (N % 16 == 0)
            float bvv = 0.0f;
            if (bias)  bvv += bias[nj + l16];
            if (bias2) bvv += bias2[nj + l16];
            #pragma unroll
            for (int i = 0; i < 8; ++i) {
                const int mr = m0 + rb + i;
                float v = acc[j][i] + bvv;
                if (add) v += add[mr * ldadd + nj + l16];
                if (lengths) v = (lengths[mr] > t) ? v : 0.0f;
                out[mr * ldc + nj + l16] = v;
            }
        }
    }
}

// ---------------------------------------------------------------------------
// embseq[t][b][e] = emb[captions[b][t]][e]     (one-time gather)
// ---------------------------------------------------------------------------
__global__ void embed_gather(const float* __restrict__ emb,
                             const int* __restrict__ captions,
                             float* __restrict__ embseq, int n)
{
    int idx = blockIdx.x * blockDim.x + threadIdx.x;
    if (idx >= n) return;
    int e   = idx % Ee;
    int rem = idx / Ee;
    int b   = rem % Bsz;
    int t   = rem / Bsz;
    embseq[idx] = emb[captions[b * Tt + t] * Ee + e];
}

// ---------------------------------------------------------------------------
// V_proj[b,k,j] = dot(spatial[b,k,:], Wv[j,:]) + bv[j]   (one-time, tiny)
// ---------------------------------------------------------------------------
__global__ __launch_bounds__(64)
void vproj_kernel(const float* __restrict__ spatial,
                  const float* __restrict__ Wv,
                  const float* __restrict__ bv,
                  float* __restrict__ Vp)
{
    __shared__ float srow[Dd];
    const int bk = blockIdx.x;   // b*Kk + k
    for (int d = threadIdx.x; d < Dd; d += 64) srow[d] = spatial[bk * Dd + d];
    __syncthreads();
    const int j = threadIdx.x;
    if (j < Kk) {
        float s = bv[j];
        const float* wr = Wv + j * Dd;
        #pragma unroll 8
        for (int d = 0; d < Dd; ++d) s += srow[d] * wr[d];
        Vp[bk * Kk + j] = s;
    }
}

// ---------------------------------------------------------------------------
// Per-step fused: LSTM nonlinearity + masked state update + spatial attention
// (z -> softmax -> context) + p = c + h.  One block per batch row.
// ---------------------------------------------------------------------------
__global__ __launch_bounds__(256)
void lstm_attn_step(const float* __restrict__ gates,
                    float* __restrict__ h, float* __restrict__ m,
                    const int* __restrict__ lengths, int t,
                    const float* __restrict__ Vp,
                    const float* __restrict__ Wg, const float* __restrict__ bg,
                    const float* __restrict__ wh, const float* __restrict__ bh_att,
                    const float* __restrict__ spatial,
                    float* __restrict__ p)
{
    __shared__ float sh[Dd];
    __shared__ float hg[Kk];
    __shared__ float alpha[Kk];
    __shared__ float swh[Kk];

    const int b   = blockIdx.x;
    const int tid = threadIdx.x;
    const bool active = lengths[b] > t;

    // LSTM cell (gate order i,f,g,o per torch LSTMCell / jnp.split)
    for (int d = tid; d < Dd; d += 256) {
        const int gbase = b * FourD + d;
        float gi = gates[gbase];
        float gf = gates[gbase + Dd];
        float gg = gates[gbase + 2 * Dd];
        float go = gates[gbase + 3 * Dd];
        float iv = sigmoidf_(gi);
        float fv = sigmoidf_(gf);
        float ov = sigmoidf_(go);
        float gv = tanhf(gg);
        float mo = m[b * Dd + d], ho = h[b * Dd + d];
        float mn = fv * mo + iv * gv;
        float hn = ov * tanhf(mn);
        float hv = active ? hn : ho;
        float mv = active ? mn : mo;
        h[b * Dd + d] = hv;
        m[b * Dd + d] = mv;
        sh[d] = hv;
    }
    if (tid < Kk) swh[tid] = wh[tid];
    __syncthreads();

    // hg[k] = Wg[k,:] . h + bg[k]
    for (int k = tid; k < Kk; k += 256) {
        float s = bg[k];
        const float* wr = Wg + k * Dd;
        #pragma unroll 8
        for (int j = 0; j < Dd; ++j) s += sh[j] * wr[j];
        hg[k] = s;
    }
    __syncthreads();

    // z[k] = sum_j wh[j]*tanh(Vp[b,k,j] + hg[j]) + bh
    for (int k = tid; k < Kk; k += 256) {
        float s = bh_att[0];
        const float* vr = Vp + (b * Kk + k) * Kk;
        for (int j = 0; j < Kk; ++j) s += swh[j] * tanhf(vr[j] + hg[j]);
        alpha[k] = s;   // z stored in alpha temporarily
    }
    __syncthreads();

    // softmax over K=49 (tiny; single lane)
    if (tid == 0) {
        float mx = alpha[0];
        for (int k2 = 1; k2 < Kk; ++k2) mx = fmaxf(mx, alpha[k2]);
        float s = 0.0f;
        for (int k2 = 0; k2 < Kk; ++k2) { float e = __expf(alpha[k2] - mx); alpha[k2] = e; s += e; }
        float inv = 1.0f / s;
        for (int k2 = 0; k2 < Kk; ++k2) alpha[k2] *= inv;
    }
    __syncthreads();

    // p = context + h
    for (int d = tid; d < Dd; d += 256) {
        float c = 0.0f;
        for (int k2 = 0; k2 < Kk; ++k2) c += alpha[k2] * spatial[(b * Kk + k2) * Dd + d];
        p[b * Dd + d] = c + sh[d];
    }
}

// ---------------------------------------------------------------------------
extern "C" void kernel_launch(void* const* d_in, const int* in_sizes, int n_in,
                              void* d_out, int out_size, void* d_ws, size_t ws_size,
                              hipStream_t stream)
{
    const float* spatial  = (const float*)d_in[0];
    const float* gfeats   = (const float*)d_in[1];
    const int*   captions = (const int*)  d_in[2];
    const int*   lengths  = (const int*)  d_in[3];
    const float* emb      = (const float*)d_in[4];
    const float* W_init_h = (const float*)d_in[5];
    const float* b_init_h = (const float*)d_in[6];
    const float* W_init_m = (const float*)d_in[7];
    const float* b_init_m = (const float*)d_in[8];
    const float* W_ih     = (const float*)d_in[9];
    const float* b_ih     = (const float*)d_in[10];
    const float* W_hh     = (const float*)d_in[11];
    const float* b_hh     = (const float*)d_in[12];
    const float* Wv       = (const float*)d_in[13];
    const float* bv       = (const float*)d_in[14];
    const float* Wg       = (const float*)d_in[15];
    const float* bg       = (const float*)d_in[16];
    const float* wh       = (const float*)d_in[17];
    const float* bh_att   = (const float*)d_in[18];
    const float* Wp       = (const float*)d_in[19];
    const float* bp       = (const float*)d_in[20];
    float* out = (float*)d_out;

    // Workspace carve-out (~6.7 MB of fp32)
    float* ws     = (float*)d_ws;
    float* h      = ws;                      // B*D
    float* mst    = h      + Bsz * Dd;       // B*D
    float* G0     = mst    + Bsz * Dd;       // B*4D  (loop-invariant gate part)
    float* gates  = G0     + Bsz * FourD;    // B*4D
    float* p      = gates  + Bsz * FourD;    // B*D   (c + h)
    float* embseq = p      + Bsz * Dd;       // T*B*E
    float* Vp     = embseq + Tt * Bsz * Ee;  // B*K*K

    const dim3 w32(32);
    const int gy = Bsz / 16;                 // 8 row-tiles

    // h0 = gfeats @ W_init_h^T + b_init_h ; m0 likewise
    wmma_gemm_f32<<<dim3((Dd + 63) / 64, gy), w32, 0, stream>>>(
        gfeats, Dd, W_init_h, Dd, b_init_h, nullptr, nullptr, 0,
        h, Dd, Dd, Dd, nullptr, 0);
    wmma_gemm_f32<<<dim3((Dd + 63) / 64, gy), w32, 0, stream>>>(
        gfeats, Dd, W_init_m, Dd, b_init_m, nullptr, nullptr, 0,
        mst, Dd, Dd, Dd, nullptr, 0);

    // G0 = gfeats @ W_ih[:, E:]^T + b_ih + b_hh   (loop-invariant)
    wmma_gemm_f32<<<dim3((FourD + 63) / 64, gy), w32, 0, stream>>>(
        gfeats, Dd, W_ih + Ee, EDsum, b_ih, b_hh, nullptr, 0,
        G0, FourD, Dd, FourD, nullptr, 0);

    // Embedding gather (all timesteps) and V_proj precompute
    const int n_eg = Tt * Bsz * Ee;
    embed_gather<<<(n_eg + 255) / 256, 256, 0, stream>>>(emb, captions, embseq, n_eg);
    vproj_kernel<<<Bsz * Kk, 64, 0, stream>>>(spatial, Wv, bv, Vp);

    // Sequential scan over T
    for (int t = 0; t < Tt; ++t) {
        // gates = G0 + emb_t @ W_ih[:, :E]^T
        wmma_gemm_f32<<<dim3((FourD + 63) / 64, gy), w32, 0, stream>>>(
            embseq + t * Bsz * Ee, Ee, W_ih, EDsum,
            nullptr, nullptr, G0, FourD, gates, FourD, Ee, FourD, nullptr, 0);
        // gates += h @ W_hh^T
        wmma_gemm_f32<<<dim3((FourD + 63) / 64, gy), w32, 0, stream>>>(
            h, Dd, W_hh, Dd,
            nullptr, nullptr, gates, FourD, gates, FourD, Dd, FourD, nullptr, 0);
        // LSTM + attention + p = c + h
        lstm_attn_step<<<Bsz, 256, 0, stream>>>(
            gates, h, mst, lengths, t, Vp, Wg, bg, wh, bh_att, spatial, p);
        // logits[:, t, :] = mask(p @ Wp^T + bp)   (157 blocks cover N=10000)
        wmma_gemm_f32<<<dim3((Vv + 63) / 64, gy), w32, 0, stream>>>(
            p, Dd, Wp, Dd, bp, nullptr, nullptr, 0,
            out + t * Vv, Tt * Vv, Dd, Vv, lengths, t);
    }
}